// S_E_29755533426928
// MI455X (gfx1250) — compile-verified
//
#include <hip/hip_runtime.h>
#include <math.h>

// int4 as a true clang ext-vector so __builtin_nontemporal_load vectorizes to
// a single global_load_b128 with the NT temporal hint.
typedef int v4i __attribute__((ext_vector_type(4)));

// ---------------------------------------------------------------------------
// Edge kernel first in the file so the disasm snippet shows it.
//
// Streams src/dst with NT b128 loads (256 MB touched once -> don't pollute
// L2; the ~24 MB of node arrays stay L2-resident), gathers node state from
// L2, early-exits non-contributing edges (~85%), and scatter-adds
// log(1 - s*i) into acc[src] with a guaranteed hardware f32 global atomic.
//
// Masks derived directly from the ORIGINAL E (before relu):
//   susceptible(src) <=> E[src] == +inf      (relu(inf-1) == inf)
//   infective(dst)   <=> E[dst] <= 1.0f      (relu(E-1) == 0)
//
// NOTE: the atomic asm deliberately has NO "memory" clobber: acc is written
// only through this asm and never read in this kernel, so the compiler is
// free to software-pipeline the independent L2 gathers / v_log_f32 of later
// edges above earlier atomics. asm volatile still pins one atomic per
// contributing edge. Completion is covered by S_ENDPGM's implicit wait-idle
// before the dependent finalize kernel runs.
// ---------------------------------------------------------------------------
__global__ void epi_edge_kernel(const int*   __restrict__ src,
                                const int*   __restrict__ dst,
                                const float* __restrict__ E,
                                const float* __restrict__ susc,
                                const float* __restrict__ infv,
                                float*       __restrict__ acc,
                                int n_edges) {
    long long t    = (long long)blockIdx.x * blockDim.x + threadIdx.x;
    long long base = t * 4;
    if (base >= n_edges) return;

    // Keep the index stream ahead of the gather-dependent branches.
    // 4096 ints = 16 KB ahead. Emits global_prefetch_b8 (gfx1250 path).
    long long pf = base + 4096;
    if (pf + 4 <= n_edges) {
        __builtin_prefetch(src + pf, 0, 1);
        __builtin_prefetch(dst + pf, 0, 1);
    }

    int s[4], d[4];
    int cnt;
    if (base + 4 <= n_edges) {
        // Non-temporal 128-bit streaming loads of 4 edges' indices.
        v4i sv = __builtin_nontemporal_load((const v4i*)(src + base));
        v4i dv = __builtin_nontemporal_load((const v4i*)(dst + base));
        s[0] = sv.x; s[1] = sv.y; s[2] = sv.z; s[3] = sv.w;
        d[0] = dv.x; d[1] = dv.y; d[2] = dv.z; d[3] = dv.w;
        cnt = 4;
    } else {
        cnt = (int)(n_edges - base);
        for (int k = 0; k < cnt; ++k) {
            s[k] = src[base + k];
            d[k] = dst[base + k];
        }
    }

#pragma unroll
    for (int k = 0; k < 4; ++k) {
        if (k >= cnt) break;
        int si = s[k];
        float es = E[si];                     // L2-resident gather
        if (!__builtin_isinf(es)) continue;   // src not susceptible -> log(1)=0
        int di = d[k];
        float ed = E[di];                     // L2-resident gather
        if (!(ed <= 1.0f)) continue;          // dst not infective -> log(1)=0
        float p = susc[si] * infv[di];        // in [0, 0.1): log well-conditioned
        float v = __logf(1.0f - p);
        // Non-returning GLOBAL_ATOMIC_ADD_F32, device scope: RMW resolves at
        // the L2 atomic units, visible across all WGPs.
        float* p_acc = acc + si;
        asm volatile("global_atomic_add_f32 %0, %1, off scope:SCOPE_DEV"
                     :
                     : "v"(p_acc), "v"(v));
    }
}

// ---------------------------------------------------------------------------
// Zero the per-node log-probability accumulator (lives in d_out).
// ---------------------------------------------------------------------------
__global__ void epi_zero_acc(float* __restrict__ acc, int n) {
    int i = blockIdx.x * blockDim.x + threadIdx.x;
    if (i < n) acc[i] = 0.0f;
}

// ---------------------------------------------------------------------------
// Per-node finalize, in place on d_out (which holds row_logp).
//   dI_prob = 1 - exp(row_logp);  E_new = (rand < dI_prob) ? incubation
//                                                          : relu(E - 1)
// ---------------------------------------------------------------------------
__global__ void epi_finalize(const float* __restrict__ E,
                             const float* __restrict__ incub,
                             const float* __restrict__ rnd,
                             float*       __restrict__ out,
                             int n) {
    int i = blockIdx.x * blockDim.x + threadIdx.x;
    if (i >= n) return;
    float logp  = out[i];
    float prob  = 1.0f - __expf(logp);
    float Erelu = fmaxf(E[i] - 1.0f, 0.0f);   // inf stays inf
    out[i] = (rnd[i] < prob) ? incub[i] : Erelu;
}

// ---------------------------------------------------------------------------
// Host launcher. Input order (from setup_inputs):
//   0: E [N] f32, 1: susceptiveness [N] f32, 2: infectiveness [N] f32,
//   3: incubation [N] f32, 4: rand_vals [N] f32,
//   5: src [N_EDGES] i32, 6: dst [N_EDGES] i32
// d_out: N floats. d_ws unused (accumulator lives in d_out).
// ---------------------------------------------------------------------------
extern "C" void kernel_launch(void* const* d_in, const int* in_sizes, int n_in,
                              void* d_out, int out_size, void* d_ws, size_t ws_size,
                              hipStream_t stream) {
    (void)n_in; (void)d_ws; (void)ws_size; (void)out_size;

    const float* E     = (const float*)d_in[0];
    const float* susc  = (const float*)d_in[1];
    const float* infv  = (const float*)d_in[2];
    const float* incub = (const float*)d_in[3];
    const float* rnd   = (const float*)d_in[4];
    const int*   src   = (const int*)d_in[5];
    const int*   dst   = (const int*)d_in[6];
    float*       out   = (float*)d_out;

    const int n       = in_sizes[0];   // nodes
    const int n_edges = in_sizes[5];   // edges

    const int BT = 256;                // 8 wave32 per block
    int node_blocks = (n + BT - 1) / BT;

    epi_zero_acc<<<node_blocks, BT, 0, stream>>>(out, n);

    long long quads = ((long long)n_edges + 3) / 4;
    int edge_blocks = (int)((quads + BT - 1) / BT);
    epi_edge_kernel<<<edge_blocks, BT, 0, stream>>>(src, dst, E, susc, infv,
                                                    out, n_edges);

    epi_finalize<<<node_blocks, BT, 0, stream>>>(E, incub, rnd, out, n);
}